// multi_loss_45062796869702
// MI455X (gfx1250) — compile-verified
//
#include <hip/hip_runtime.h>
#include <hip/hip_bf16.h>

#define B_N     8192
#define TILE    1024          // j-tile staged in LDS per block
#define JCHUNKS 8             // j split across blocks
#define IBLOCKS (B_N / 256)   // 32 blocks cover i, one i per thread
#define NBLK    (IBLOCKS * JCHUNKS)  // 256 blocks total

typedef __attribute__((ext_vector_type(2))) float v2f;
typedef __attribute__((ext_vector_type(8))) float v8f;

// ---------------------------------------------------------------------------
// Kernel 1: pairwise rank-loss partial sums.
// blockIdx -> (i-chunk, j-chunk). Each thread owns one i; the block's j-range
// (1024 elems of t and p) is staged into LDS via CDNA5 async global->LDS
// loads (one b128 per lane per array), then each thread walks the tile.
// Exact reference semantics including the dt==0 tie case (|dp|).
// Deterministic: fixed LDS tree reduction, no float atomics.
// ---------------------------------------------------------------------------
__global__ __launch_bounds__(256) void pair_kernel(const float* __restrict__ t,
                                                   const float* __restrict__ p,
                                                   float* __restrict__ partials) {
    __shared__ float sh_t[TILE];
    __shared__ float sh_p[TILE];
    __shared__ float red[256];

    const int tid  = threadIdx.x;
    const int iblk = blockIdx.x % IBLOCKS;
    const int jblk = blockIdx.x / IBLOCKS;

    const int   i  = iblk * 256 + tid;
    const float ti = t[i];
    const float pi = p[i];

    const int j0 = jblk * TILE;

    // --- async stage of the j-tile into LDS (16B per lane per array) ---
    {
        unsigned lds_t_off = (unsigned)(size_t)(&sh_t[tid * 4]);
        unsigned lds_p_off = (unsigned)(size_t)(&sh_p[tid * 4]);
        const float* gt = t + j0 + tid * 4;
        const float* gp = p + j0 + tid * 4;
        asm volatile("global_load_async_to_lds_b128 %0, %1, off"
                     :: "v"(lds_t_off), "v"(gt) : "memory");
        asm volatile("global_load_async_to_lds_b128 %0, %1, off"
                     :: "v"(lds_p_off), "v"(gp) : "memory");
        asm volatile("s_wait_asynccnt 0" ::: "memory");
    }
    __syncthreads();   // every wave waited on its own ASYNCcnt before arriving

    // --- 1024 pair losses for this thread's i ---
    float acc = 0.0f;
#pragma unroll 8
    for (int j = 0; j < TILE; ++j) {
        const float dt = ti - sh_t[j];
        const float dp = pi - sh_p[j];
        // dt>0 -> relu(-dp); dt<0 -> relu(dp); dt==0 -> |dp| (ties, and 0 on diag)
        const float l = (dt > 0.0f) ? fmaxf(0.0f, -dp)
                      : ((dt < 0.0f) ? fmaxf(0.0f, dp) : fabsf(dp));
        acc += l;
    }

    // --- deterministic block reduction ---
    red[tid] = acc;
    __syncthreads();
    for (int s = 128; s > 0; s >>= 1) {
        if (tid < s) red[tid] += red[tid + s];
        __syncthreads();
    }
    if (tid == 0) partials[blockIdx.x] = red[0];
}

// ---------------------------------------------------------------------------
// Kernel 2: O(B) terms + final combine. Single block, deterministic.
// The 256 rank partials are reduced 256->64 with fixed-order adds, then
// 64->16 with one exact f32 WMMA (A = ones 16x4, so D row0 = column sums of
// B; row 0 lives in VGPR0 lanes 0..15 per the documented C/D layout), then
// 16->1 serially in thread 0.
// ---------------------------------------------------------------------------
__global__ __launch_bounds__(256) void finalize_kernel(const float* __restrict__ t,
                                                       const float* __restrict__ p,
                                                       const float* __restrict__ partials,
                                                       float* __restrict__ out) {
    __shared__ float red[256];
    __shared__ float s64[64];
    __shared__ float cres[16];

    const int tid = threadIdx.x;

    // ---- mse and cosine sums over 8192 elements ----
    float msum = 0.0f, csum = 0.0f;
    for (int i = tid; i < B_N; i += 256) {
        const float a = t[i];
        const float b = p[i];
        const float d = a - b;
        msum += d * d;
        csum += (a * b) / fmaxf(fabsf(a) * fabsf(b), 1e-8f);
    }

    red[tid] = msum;
    __syncthreads();
    for (int s = 128; s > 0; s >>= 1) {
        if (tid < s) red[tid] += red[tid + s];
        __syncthreads();
    }
    const float mse_total = red[0];
    __syncthreads();

    red[tid] = csum;
    __syncthreads();
    for (int s = 128; s > 0; s >>= 1) {
        if (tid < s) red[tid] += red[tid + s];
        __syncthreads();
    }
    const float cos_total = red[0];
    __syncthreads();

    // ---- rank partials: 256 -> 64 (fixed order) ----
    if (tid < 64) {
        s64[tid] = ((partials[tid] + partials[tid + 64]) +
                    (partials[tid + 128] + partials[tid + 192]));
    }
    __syncthreads();

    // ---- 64 -> 16 via one exact f32 WMMA on wave 0 (EXEC all ones) ----
    if (tid < 32) {
        v2f a;  a.x = 1.0f;      a.y = 1.0f;          // A = ones(16x4)
        v2f b;  b.x = s64[tid];  b.y = s64[tid + 32]; // any bijection of the 64
                                                      // values onto B slots is
                                                      // fine: we only want the
                                                      // total of all columns
        v8f c = {};
        c = __builtin_amdgcn_wmma_f32_16x16x4_f32(
                /*neg_a=*/false, a, /*neg_b=*/false, b,
                /*c_mod=*/(short)0, c, /*reuse_a=*/false, /*reuse_b=*/false);
        if (tid < 16) cres[tid] = c[0];   // D row 0 = 16 column sums
    }
    __syncthreads();

    if (tid == 0) {
        float rank_total = 0.0f;
        for (int k = 0; k < 16; ++k) rank_total += cres[k];
        rank_total *= 0.5f;   // upper triangle of the symmetric pair matrix

        const float mse     = mse_total / (float)B_N;
        const float simloss = fmaxf(0.0f, 1.0f - cos_total / (float)B_N);
        out[0] = 1.0f * mse + 1.0f * simloss + 1.0f * rank_total;
    }
}

// ---------------------------------------------------------------------------
extern "C" void kernel_launch(void* const* d_in, const int* in_sizes, int n_in,
                              void* d_out, int out_size, void* d_ws, size_t ws_size,
                              hipStream_t stream) {
    const float* y_trues = (const float*)d_in[0];
    const float* y_preds = (const float*)d_in[1];
    float*       out     = (float*)d_out;
    float*       wsf     = (float*)d_ws;   // 256 partials (1 KB)

    pair_kernel<<<NBLK, 256, 0, stream>>>(y_trues, y_preds, wsf);
    finalize_kernel<<<1, 256, 0, stream>>>(y_trues, y_preds, wsf, out);
}